// UniversalSoap_79757542687242
// MI455X (gfx1250) — compile-verified
//
#include <hip/hip_runtime.h>
#include <math.h>

// CDNA5 / gfx1250: wave32, WMMA f32 16x16x4.
typedef __attribute__((ext_vector_type(2))) float v2f;
typedef __attribute__((ext_vector_type(8))) float v8f;

// 4-way lane-varying select -> 3 v_cndmask
__device__ __forceinline__ float sel4(int k, float a, float b, float c, float d) {
    float lo = (k & 1) ? b : a;
    float hi = (k & 1) ? d : c;
    return (k & 2) ? hi : lo;
}

// One rank-4 update: 4 points -> one V_WMMA_F32_16X16X4_F32 into acc.
// GUARD=false version is fully branch-free (no exec manipulation).
template <bool GUARD>
__device__ __forceinline__ void soap_chunk(
    const float* __restrict__ coo, const int* __restrict__ numbers,
    int base, int N, int half, int hi2, int lo2, v8f& acc)
{
    v2f a, b;
    #pragma unroll
    for (int j = 0; j < 2; ++j) {
        const int q = base + 2 * half + j;   // K slot = 2*half + j
        float x, y, z;
        int spec;
        if (!GUARD || q < N) {
            x = coo[3 * q + 0];
            y = coo[3 * q + 1];
            z = coo[3 * q + 2];
            spec = numbers[q];
        } else {
            x = y = z = 0.0f;
            spec = -1;                       // zero A column -> no contribution
        }
        // radial part (UNIT = RC/6 = 1.0)
        const float r2 = x * x + y * y + z * z;
        const float d  = sqrtf(r2);
        const float tc = 1.0f - d * (1.0f / 6.0f);
        const float cut = (d < 6.0f) ? tc * tc : 0.0f;
        const float rad = __expf(-0.5f * r2) * cut;
        const float f1 = rad * r2, f2 = f1 * r2, f3 = f2 * r2;
        const float fa = sel4(lo2, rad, f1, f2, f3);
        const float av = (spec == hi2) ? fa : 0.0f;

        // regular solid harmonics grid Y[l1][l2], L = 4 (closed form)
        const float xx = x * x, yy = y * y;
        const float zz5 = 5.0f * z * z;
        const float Y00 = 1.0f;
        const float Y01 = 0.5f * y;
        const float Y02 = 0.25f * x * y;
        const float Y03 = (3.0f * xx * y - yy * y) * (1.0f / 48.0f);
        const float Y10 = 0.5f * x;
        const float Y11 = z;
        const float Y12 = 0.5f * y * z;
        const float Y13 = 0.25f * x * y * z;
        const float Y20 = 0.125f * (xx - yy);
        const float Y21 = 0.5f * x * z;
        const float Y22 = 0.25f * (3.0f * z * z - r2);
        const float Y23 = y * (zz5 - r2) * (1.0f / 16.0f);
        const float Y30 = (xx * x - 3.0f * x * yy) * (1.0f / 48.0f);
        const float Y31 = 0.125f * z * (xx - yy);
        const float Y32 = x * (zz5 - r2) * (1.0f / 16.0f);
        const float Y33 = z * (zz5 - 3.0f * r2) * (1.0f / 12.0f);

        float bv = sel4(hi2,
                        sel4(lo2, Y00, Y01, Y02, Y03),
                        sel4(lo2, Y10, Y11, Y12, Y13),
                        sel4(lo2, Y20, Y21, Y22, Y23),
                        sel4(lo2, Y30, Y31, Y32, Y33));
        if (GUARD && q >= N) bv = 0.0f;

        a[j] = av;
        b[j] = bv;
    }
    // D = A(16x4,f32) * B(4x16,f32) + C ; accumulator chain (D->C) has no hazard
    acc = __builtin_amdgcn_wmma_f32_16x16x4_f32(
        /*neg_a=*/false, a, /*neg_b=*/false, b,
        /*c_mod=*/(short)0, acc, /*reuse_a=*/false, /*reuse_b=*/false);
}

// Each wave accumulates a 16x16 f32 tile C[M=spec*4+n][N=l1*4+l2] and writes
// its partial tile (256 floats) to ws[wid*256 ...]. Deterministic (no atomics).
__global__ void __launch_bounds__(256) soap_accum_kernel(
    const float* __restrict__ coo, const int* __restrict__ numbers,
    float* __restrict__ partial, int N, int W)
{
    const int lane = threadIdx.x & 31;
    const int wid  = (int)((blockIdx.x * blockDim.x + threadIdx.x) >> 5);
    const int half = lane >> 4;       // which K pair this lane holds
    const int sub  = lane & 15;       // A-row (M) == B-col (N) index for this lane
    const int hi2  = sub >> 2;        // A: species row   | B: grid row l1
    const int lo2  = sub & 3;         // A: radial n      | B: grid col l2

    v8f acc = {};                     // 16x16 f32 accumulator (8 VGPRs)
    const int stride = W * 4;
    int base = wid * 4;

    // Branch-free steady state, 2 WMMAs per iteration.
    for (; base + stride + 4 <= N; base += 2 * stride) {
        __builtin_prefetch(coo + (size_t)3 * (size_t)(base + 2 * stride), 0, 0);
        soap_chunk<false>(coo, numbers, base,          N, half, hi2, lo2, acc);
        soap_chunk<false>(coo, numbers, base + stride, N, half, hi2, lo2, acc);
    }
    if (base + 4 <= N) {              // one leftover full chunk
        soap_chunk<false>(coo, numbers, base, N, half, hi2, lo2, acc);
        base += stride;
    }
    if (base < N) {                   // at most one partial (guarded) chunk
        soap_chunk<true>(coo, numbers, base, N, half, hi2, lo2, acc);
    }

    // Write full partial tile: lane(0-15) vgpr j -> M=j ; lane(16-31) -> M=j+8
    float* pb = partial + (size_t)wid * 256;
    #pragma unroll
    for (int j = 0; j < 8; ++j)
        pb[(j + 8 * half) * 16 + sub] = acc[j];
}

__device__ __forceinline__ float acoef(int n, int l) {
    const float fact[8] = {1.f, 1.f, 2.f, 6.f, 24.f, 120.f, 720.f, 5040.f};
    return 1.0f / ((float)(2 * l + 1) * (float)(1 << (2 * n + l)) * fact[n] * fact[n + l]);
}

// Single block: deterministically reduce W partial tiles, build p[1024],
// scale by nnl, L2-normalize, write out.
__global__ void __launch_bounds__(256) soap_finalize_kernel(
    const float* __restrict__ partial, float* __restrict__ out, int W)
{
    __shared__ float c[256];
    __shared__ float p[1024];
    __shared__ float red[256];
    const int t = threadIdx.x;

    float s = 0.0f;
    for (int w = 0; w < W; ++w) s += partial[(size_t)w * 256 + t];
    c[t] = s;            // c[spec][n][l1][l2] = c[(spec*4+n)*16 + l1*4 + l2]
    __syncthreads();

    for (int e = t; e < 1024; e += 256) {
        const int l  = e & 3;
        const int n2 = (e >> 2) & 3;
        const int n1 = (e >> 4) & 3;
        const int s2 = (e >> 6) & 3;
        const int s1 = (e >> 8) & 3;
        const float* cA = c + (s2 * 4 + n2) * 16;  // c[None,:,None] factor
        const float* cB = c + (s1 * 4 + n1) * 16;  // c[:,None,:,None] factor
        float v = 0.0f;
        #pragma unroll
        for (int k = 0; k < 4; ++k) {
            const float yr = (k < l) ? 2.0f : ((k == l) ? 1.0f : 0.0f); // Yr[l][k]
            const float yi = (k < l) ? 2.0f : 0.0f;                     // Yi[k][l]
            v += cA[l * 4 + k] * cB[l * 4 + k] * yr;   // sum over last axis
            v += cA[k * 4 + l] * cB[k * 4 + l] * yi;   // sum over axis -2
        }
        v *= sqrtf(acoef(n1, l) * acoef(n2, l));       // nnl[n1][n2][l]
        p[e] = v;
    }
    __syncthreads();

    float ss = 0.0f;
    for (int e = t; e < 1024; e += 256) ss += p[e] * p[e];
    red[t] = ss;
    __syncthreads();
    for (int off = 128; off > 0; off >>= 1) {
        if (t < off) red[t] += red[t + off];
        __syncthreads();
    }
    const float inv = 1.0f / (sqrtf(red[0]) + 1.1920929e-07f);
    for (int e = t; e < 1024; e += 256) out[e] = p[e] * inv;
}

extern "C" void kernel_launch(void* const* d_in, const int* in_sizes, int n_in,
                              void* d_out, int out_size, void* d_ws, size_t ws_size,
                              hipStream_t stream) {
    const float* coo     = (const float*)d_in[0];
    const int*   numbers = (const int*)d_in[1];
    float*       out     = (float*)d_out;
    float*       partial = (float*)d_ws;

    const int N = in_sizes[1];   // number of points (coo is 3N floats)

    // 8 waves / 256-thread block; each wave needs 256 floats (1 KiB) of ws.
    size_t maxBlocksByWs = ws_size / (size_t)(8 * 256 * sizeof(float));
    int blocks = (int)(maxBlocksByWs < 128 ? (maxBlocksByWs ? maxBlocksByWs : 1) : 128);
    const int threads = 256;
    const int W = blocks * (threads / 32);

    soap_accum_kernel<<<blocks, threads, 0, stream>>>(coo, numbers, partial, N, W);
    soap_finalize_kernel<<<1, 256, 0, stream>>>(partial, out, W);
}